// AdaptiveTokenSampling_86131274154442
// MI455X (gfx1250) — compile-verified
//
#include <hip/hip_runtime.h>
#include <math.h>

#define B   8
#define H   12
#define N   1024
#define D   64
#define K   256
#define NM1 1023
#define EPSF 1e-6f
#define MASK_VALUE (-1.7014117e38f) /* -FLT_MAX/2 */

typedef __attribute__((ext_vector_type(2))) float v2f;
typedef __attribute__((ext_vector_type(8))) float v8f;

// GCC-style int4 to match the async-LDS builtin prototypes (b128 <-> v4i)
typedef int v4i __attribute__((vector_size(16)));
typedef __attribute__((address_space(1))) v4i* gv4i_ptr;
typedef __attribute__((address_space(3))) v4i* lv4i_ptr;

// ---------------------------------------------------------------------------
// Kernel 1: score[b,s] = sum_h attn[b,h,0,s+1] * ||value[b,h,s+1,:]||
// One wave per (b, 16-token tile). Sum-of-squares over d=64 on the matrix
// pipe: chained 16x V_WMMA_F32_16X16X4_F32 with A == ones(16x4), so
// D[m,n] = sum_k (v^2)[k,n]: every row of D equals the column sums, making
// the result insensitive to the exact K-slot striping of B; we only rely on
// the documented "B columns stripe across lanes" (N = lane & 15).
// Out-of-range lanes (only column 15 of the last tile) are handled by
// CLAMPING the token index -- loads stay unconditional straight-line b64,
// no EXEC churn, no per-load waits; the garbage column is never stored.
// ---------------------------------------------------------------------------
__global__ __launch_bounds__(32) void ats_scores(const float* __restrict__ attn,
                                                 const float* __restrict__ value,
                                                 float* __restrict__ scores) {
  const int lane  = threadIdx.x & 31;
  const int b     = blockIdx.x >> 6;   // 64 tiles per batch
  const int tile  = blockIdx.x & 63;
  const int tok16 = lane & 15;
  const int s     = tile * 16 + tok16;
  const bool valid = (s < NM1);
  const int sc    = valid ? s : (NM1 - 1);  // clamp: in-bounds, result discarded
  const int token = sc + 1;                 // <= 1023 < N
  const int dpair = 2 * (lane >> 4);        // lanes 0-15: d%4 in {0,1}; 16-31: {2,3}

  v2f a; a.x = 1.0f; a.y = 1.0f;            // A = ones(16x4), hoisted

  float acc = 0.f;
  for (int h = 0; h < H; ++h) {
    const float* vrow = value + (((size_t)(b * H + h) * N + token) * D) + dpair;
    float normsq;
#if __has_builtin(__builtin_amdgcn_wmma_f32_16x16x4_f32)
    v8f c = {0.f, 0.f, 0.f, 0.f, 0.f, 0.f, 0.f, 0.f};
#pragma unroll
    for (int kk = 0; kk < 16; ++kk) {       // chain K: 16 * 4 = 64 = D
      const v2f p = *(const v2f*)(vrow + kk * 4);  // unconditional b64 load
      v2f bm; bm.x = p.x * p.x; bm.y = p.y * p.y;  // B column = v^2 of this token
      c = __builtin_amdgcn_wmma_f32_16x16x4_f32(
              /*neg_a=*/false, a, /*neg_b=*/false, bm,
              /*c_mod=*/(short)0, c, /*reuse_a=*/false, /*reuse_b=*/false);
    }
    normsq = c[0];                          // all rows of D identical = column sum
#else
    normsq = 0.f;
#pragma unroll
    for (int d0 = 0; d0 < D; d0 += 2) {
      const v2f p = *(const v2f*)(value + (((size_t)(b * H + h) * N + token) * D) + d0);
      normsq += p.x * p.x + p.y * p.y;
    }
#endif
    const float cls = attn[((size_t)(b * H + h) * N) * N + token];  // clamped, in-bounds
    acc += cls * sqrtf(normsq);
  }
  if (lane < 16 && valid) scores[b * NM1 + s] = acc;
}

// ---------------------------------------------------------------------------
// Kernel 2: pseudo_logits[b,s] = log(score/(sum+eps) + eps), masked fill
// ---------------------------------------------------------------------------
__global__ __launch_bounds__(256) void ats_logits(const float* __restrict__ scores,
                                                  const int* __restrict__ mask,
                                                  float* __restrict__ logits) {
  const int b = blockIdx.x;
  const int t = threadIdx.x;
  __shared__ float red[256];
  float ls = 0.f;
  for (int s = t; s < NM1; s += 256) ls += scores[b * NM1 + s];
  red[t] = ls;
  __syncthreads();
  for (int off = 128; off > 0; off >>= 1) {
    if (t < off) red[t] += red[t + off];
    __syncthreads();
  }
  const float denom = red[0] + EPSF;
  for (int s = t; s < NM1; s += 256) {
    float lg = logf(scores[b * NM1 + s] / denom + EPSF);
    if (mask[b * N + (s + 1)] == 0) lg = MASK_VALUE;
    logits[b * NM1 + s] = lg;
  }
}

// ---------------------------------------------------------------------------
// Kernel 3: sampled[b,ki] = argmax_s(logits[b,s] + gumbel[b,ki,s]) + 1
// One wave per (b,ki); first-index tie-break (matches jnp.argmax).
// ---------------------------------------------------------------------------
__global__ __launch_bounds__(256) void ats_sample(const float* __restrict__ logits,
                                                  const float* __restrict__ gu,
                                                  int* __restrict__ sampled) {
  const int gw   = (blockIdx.x * blockDim.x + threadIdx.x) >> 5;
  const int lane = threadIdx.x & 31;
  const int b    = gw >> 8;     // gw / K
  const int ki   = gw & 255;    // gw % K
  const float* lrow = logits + b * NM1;
  const float* grow = gu + (size_t)(b * K + ki) * NM1;
  float best = -3.4e38f;
  int   bidx = 0;
  for (int s = lane; s < NM1; s += 32) {
    float u = grow[s];
    float g = -logf(-logf(u + EPSF) + EPSF);  // gumbel = -_log(-_log(u))
    float v = lrow[s] + g;
    if (v > best) { best = v; bidx = s; }     // ascending s -> keeps first max
  }
  for (int off = 16; off > 0; off >>= 1) {
    float ov = __shfl_down(best, off, 32);
    int   oi = __shfl_down(bidx, off, 32);
    if (ov > best || (ov == best && oi < bidx)) { best = ov; bidx = oi; }
  }
  if (lane == 0) sampled[b * K + ki] = bidx + 1;
}

// ---------------------------------------------------------------------------
// Kernel 4: per-batch sort (O(k^2) rank sort), dedupe, re-sort; emit
// ids (int scratch + float d_out) and new_mask (float d_out).
// ---------------------------------------------------------------------------
__global__ __launch_bounds__(256) void ats_sortdedupe(const int* __restrict__ sampled,
                                                      int* __restrict__ ids_int,
                                                      float* __restrict__ out_mask,
                                                      float* __restrict__ out_ids) {
  const int b = blockIdx.x;
  const int t = threadIdx.x;
  __shared__ int v[K];
  __shared__ int srt[K];
  __shared__ int w[K];
  v[t] = sampled[b * K + t];
  __syncthreads();
  // stable ascending rank sort
  {
    const int my = v[t];
    int r = 0;
    for (int j = 0; j < K; ++j) { int o = v[j]; r += (o < my) || (o == my && j < t); }
    srt[r] = my;
  }
  __syncthreads();
  // dup -> 0
  w[t] = (t > 0 && srt[t] == srt[t - 1]) ? 0 : srt[t];
  __syncthreads();
  // re-sort (zeros to front), reuse srt as uniq
  {
    const int my = w[t];
    int r = 0;
    for (int j = 0; j < K; ++j) { int o = w[j]; r += (o < my) || (o == my && j < t); }
    srt[r] = my;
  }
  __syncthreads();
  const int u = srt[t];
  if (t == 0) {
    ids_int[b * 257] = 0;
    out_ids[b * 257] = 0.f;
    out_mask[b * 257] = 1.f;     // constant_values=True
  }
  ids_int[b * 257 + t + 1] = u;
  out_ids[b * 257 + t + 1] = (float)u;
  out_mask[b * 257 + t + 1] = (u != 0) ? 1.f : 0.f;
}

// ---------------------------------------------------------------------------
// Kernel 5: new_attn[b,h,t,:] = attn[b,h,ids[b,t],:]
// One wave per 4KB row. Async DMA: global -> LDS -> global, B128 per lane.
// ---------------------------------------------------------------------------
#if defined(__gfx1250__) && \
    __has_builtin(__builtin_amdgcn_global_load_async_to_lds_b128) && \
    __has_builtin(__builtin_amdgcn_global_store_async_from_lds_b128)
#define ATS_ASYNC 1
#else
#define ATS_ASYNC 0
#endif

__global__ __launch_bounds__(256) void ats_gather(const float* __restrict__ attn,
                                                  const int* __restrict__ ids_int,
                                                  float* __restrict__ out_attn) {
  const int wave = (blockIdx.x * blockDim.x + threadIdx.x) >> 5;  // exact: 24672 waves
  const int lane = threadIdx.x & 31;
  const int t  = wave % 257;
  const int bh = wave / 257;
  const int h  = bh % H;
  const int b  = bh / H;
  const int id = ids_int[b * 257 + t];
  const float* src = attn + (((size_t)(b * H + h) * N + id) * N);
  float* dst = out_attn + (size_t)wave * N;
#if ATS_ASYNC
  __shared__ __align__(16) char lds[8][4096];   // 4KB per wave
  char* lp = &lds[threadIdx.x >> 5][0];
  const char* sp = (const char*)src;
  char* dp = (char*)dst;
#pragma unroll
  for (int c = 0; c < 8; ++c) {                 // 8 x (32 lanes x 16B) = 4KB
    __builtin_amdgcn_global_load_async_to_lds_b128(
        (gv4i_ptr)(sp + c * 512 + lane * 16),
        (lv4i_ptr)(lp + c * 512 + lane * 16),
        0, 0);
  }
#if __has_builtin(__builtin_amdgcn_s_wait_asynccnt)
  __builtin_amdgcn_s_wait_asynccnt(0);
#else
  asm volatile("s_wait_asynccnt 0" ::: "memory");
#endif
#pragma unroll
  for (int c = 0; c < 8; ++c) {
    __builtin_amdgcn_global_store_async_from_lds_b128(
        (gv4i_ptr)(dp + c * 512 + lane * 16),
        (lv4i_ptr)(lp + c * 512 + lane * 16),
        0, 0);
  }
#if __has_builtin(__builtin_amdgcn_s_wait_asynccnt)
  __builtin_amdgcn_s_wait_asynccnt(0);
#else
  asm volatile("s_wait_asynccnt 0" ::: "memory");
#endif
#else
  const float4* s4 = (const float4*)src;
  float4* d4 = (float4*)dst;
#pragma unroll
  for (int c = 0; c < 8; ++c) d4[c * 32 + lane] = s4[c * 32 + lane];
#endif
}

// ---------------------------------------------------------------------------
extern "C" void kernel_launch(void* const* d_in, const int* in_sizes, int n_in,
                              void* d_out, int out_size, void* d_ws, size_t ws_size,
                              hipStream_t stream) {
  const float* attn  = (const float*)d_in[0];   // (8,12,1024,1024) f32
  const float* value = (const float*)d_in[1];   // (8,12,1024,64)   f32
  const int*   mask  = (const int*)d_in[2];     // (8,1024) bool->int
  const float* gu    = (const float*)d_in[3];   // (8,256,1023) f32
  (void)in_sizes; (void)n_in; (void)out_size; (void)ws_size;

  char* ws = (char*)d_ws;
  float* scores  = (float*)(ws + 0);        // 8*1023 f32
  float* logits  = (float*)(ws + 32768);    // 8*1023 f32
  int*   sampled = (int*)(ws + 65536);      // 8*256  i32
  int*   ids_int = (int*)(ws + 73728);      // 8*257  i32

  float* out_attn = (float*)d_out;                              // 8*12*257*1024
  float* out_mask = out_attn + (size_t)B * H * 257 * N;         // 8*257
  float* out_ids  = out_mask + B * 257;                         // 8*257

  ats_scores    <<<B * 64, 32,  0, stream>>>(attn, value, scores);
  ats_logits    <<<B,      256, 0, stream>>>(scores, mask, logits);
  ats_sample    <<<(B * K * 32) / 256, 256, 0, stream>>>(logits, gu, sampled);
  ats_sortdedupe<<<B,      256, 0, stream>>>(sampled, ids_int, out_mask, out_ids);
  ats_gather    <<<(B * H * 257 * 32) / 256, 256, 0, stream>>>(attn, ids_int, out_attn);
}